// SelfContact_80221399154693
// MI455X (gfx1250) — compile-verified
//
#include <hip/hip_runtime.h>
#include <math.h>

#define NVERT 6890
#define NFACE 13776
#define NBATCH 2
#define NJT 431            // ceil(NVERT/16)
#define NTT 861            // NFACE/16 (exact)
#define TRI_STRIDE 28      // floats per (batch,tri) record: 7 quantities x 4
#define EUCLTHRES 0.02f
#define INV_2PI 0.15915494309189535f

typedef __attribute__((ext_vector_type(2))) float v2f;
typedef __attribute__((ext_vector_type(8))) float v8f;

// D = A(16x4) * B(4x16) + C, f32, one instruction: v_wmma_f32_16x16x4_f32
__device__ __forceinline__ v8f wmma4(v2f a, v2f b, v8f c) {
  return __builtin_amdgcn_wmma_f32_16x16x4_f32(false, a, false, b, (short)0, c,
                                               false, false);
}

// raw v_sqrt_f32 (inputs are O(1) squared norms; no denormal fixup needed)
__device__ __forceinline__ float fast_sqrt(float x) {
  return __builtin_amdgcn_sqrtf(x);
}

__device__ __forceinline__ int iminh(int a, int b) { return a < b ? a : b; }

// ---------------------------------------------------------------------------
// Prep: per-vertex WMMA row/col fragments + per-triangle constant 4-vectors.
//   arow[v] = [x, y, z, 1]          (A-matrix rows, homogeneous point)
//   bcol[v] = [-2x, -2y, -2z, |v|2] (B-matrix cols for pairwise d^2)
//   tri[t]  = 7 x float4:
//     q0 [-(A+B), A.B]  q1 [-(B+C), B.C]  q2 [-(C+A), C.A]
//     q3 [-n, det0]  (n = AxB+BxC+CxA, det0 = A.(BxC))  => det = q3 . [p,1]
//     q4 [-2A, |A|2]  q5 [-2B, |B|2]  q6 [-2C, |C|2]    => |a|2 etc. (+|p|2)
// ---------------------------------------------------------------------------
__global__ void prep_kernel(const float* __restrict__ verts,
                            const int* __restrict__ faces,
                            float* __restrict__ arow,
                            float* __restrict__ bcol,
                            float* __restrict__ sqv,
                            float* __restrict__ tri) {
  int tid = blockIdx.x * blockDim.x + threadIdx.x;
  if (tid < NBATCH * NVERT) {
    const float* p = verts + (size_t)tid * 3;
    float x = p[0], y = p[1], z = p[2];
    float sq = x * x + y * y + z * z;
    reinterpret_cast<float4*>(arow)[tid] = make_float4(x, y, z, 1.0f);
    reinterpret_cast<float4*>(bcol)[tid] =
        make_float4(-2.0f * x, -2.0f * y, -2.0f * z, sq);
    sqv[tid] = sq;
  }
  if (tid < NBATCH * NFACE) {
    int b = tid / NFACE;
    int t = tid % NFACE;
    int f0 = faces[t * 3 + 0], f1 = faces[t * 3 + 1], f2 = faces[t * 3 + 2];
    const float* A = verts + ((size_t)b * NVERT + f0) * 3;
    const float* B = verts + ((size_t)b * NVERT + f1) * 3;
    const float* C = verts + ((size_t)b * NVERT + f2) * 3;
    float ax = A[0], ay = A[1], az = A[2];
    float bx = B[0], by = B[1], bz = B[2];
    float cx = C[0], cy = C[1], cz = C[2];
    float cAB = ax * bx + ay * by + az * bz;
    float cBC = bx * cx + by * cy + bz * cz;
    float cCA = cx * ax + cy * ay + cz * az;
    float abX = ay * bz - az * by, abY = az * bx - ax * bz, abZ = ax * by - ay * bx;
    float bcX = by * cz - bz * cy, bcY = bz * cx - bx * cz, bcZ = bx * cy - by * cx;
    float caX = cy * az - cz * ay, caY = cz * ax - cx * az, caZ = cx * ay - cy * ax;
    float nx = abX + bcX + caX, ny = abY + bcY + caY, nz = abZ + bcZ + caZ;
    float d0 = ax * bcX + ay * bcY + az * bcZ;
    float nA2 = ax * ax + ay * ay + az * az;
    float nB2 = bx * bx + by * by + bz * bz;
    float nC2 = cx * cx + cy * cy + cz * cz;
    float4* o = reinterpret_cast<float4*>(tri + (size_t)tid * TRI_STRIDE);
    o[0] = make_float4(-(ax + bx), -(ay + by), -(az + bz), cAB);
    o[1] = make_float4(-(bx + cx), -(by + cy), -(bz + cz), cBC);
    o[2] = make_float4(-(cx + ax), -(cy + ay), -(cz + az), cCA);
    o[3] = make_float4(-nx, -ny, -nz, d0);
    o[4] = make_float4(-2.f * ax, -2.f * ay, -2.f * az, nA2);
    o[5] = make_float4(-2.f * bx, -2.f * by, -2.f * bz, nB2);
    o[6] = make_float4(-2.f * cx, -2.f * cy, -2.f * cz, nC2);
  }
}

// ---------------------------------------------------------------------------
// Winding numbers: block = (batch, 16-point tile); 8 waves stride tri tiles.
// A frag: lane holds M=lane%16, K pair {2*(lane/16), +1}.  B frag mirrored.
// C/D: VGPR r, lane -> element (M = r + 8*(lane/16), N = lane%16).
// ---------------------------------------------------------------------------
__global__ void __launch_bounds__(256)
wn_kernel(const float* __restrict__ arow, const float* __restrict__ sqv,
          const float* __restrict__ tri, float* __restrict__ outExt) {
  const int b = blockIdx.x / NJT;
  const int jt = blockIdx.x % NJT;
  const int j0 = jt * 16;
  const int lane = threadIdx.x & 31;
  const int wave = threadIdx.x >> 5;
  const int half = lane >> 4;
  const int ln = lane & 15;

  int j = iminh(j0 + ln, NVERT - 1);
  v2f afrag =
      *reinterpret_cast<const v2f*>(arow + ((size_t)b * NVERT + j) * 4 + 2 * half);

  v8f ppc;  // |p|^2 broadcast per C/D row
#pragma unroll
  for (int r = 0; r < 8; ++r)
    ppc[r] = sqv[(size_t)b * NVERT + iminh(j0 + 8 * half + r, NVERT - 1)];
  v8f zero = {0.f, 0.f, 0.f, 0.f, 0.f, 0.f, 0.f, 0.f};

  float acc[8];
#pragma unroll
  for (int r = 0; r < 8; ++r) acc[r] = 0.f;

  const float* triB = tri + (size_t)b * NFACE * TRI_STRIDE;

  for (int tt = wave; tt < NTT; tt += 8) {
    const float* tp = triB + (size_t)(tt * 16 + ln) * TRI_STRIDE + 2 * half;
    if (tt + 8 < NTT)  // global_prefetch_b8 of next tile this wave will touch
      __builtin_prefetch(triB + (size_t)((tt + 8) * 16 + ln) * TRI_STRIDE, 0, 1);
    v2f b0 = *reinterpret_cast<const v2f*>(tp + 0);
    v2f b1 = *reinterpret_cast<const v2f*>(tp + 4);
    v2f b2 = *reinterpret_cast<const v2f*>(tp + 8);
    v2f b3 = *reinterpret_cast<const v2f*>(tp + 12);
    v2f b4 = *reinterpret_cast<const v2f*>(tp + 16);
    v2f b5 = *reinterpret_cast<const v2f*>(tp + 20);
    v2f b6 = *reinterpret_cast<const v2f*>(tp + 24);
    v8f vAB = wmma4(afrag, b0, ppc);   // a.b
    v8f vBC = wmma4(afrag, b1, ppc);   // b.c
    v8f vCA = wmma4(afrag, b2, ppc);   // c.a
    v8f vDT = wmma4(afrag, b3, zero);  // det[a,b,c]
    v8f vNA = wmma4(afrag, b4, ppc);   // |a|^2
    v8f vNB = wmma4(afrag, b5, ppc);   // |b|^2
    v8f vNC = wmma4(afrag, b6, ppc);   // |c|^2
#pragma unroll
    for (int r = 0; r < 8; ++r) {
      float na = fast_sqrt(fmaxf(vNA[r], 0.f));
      float nb = fast_sqrt(fmaxf(vNB[r], 0.f));
      float nc = fast_sqrt(fmaxf(vNC[r], 0.f));
      float denom = na * nb * nc + vAB[r] * nc + vBC[r] * na + vCA[r] * nb;
      acc[r] += atan2f(vDT[r], denom);
    }
  }

  // sum over triangle columns: butterfly within each 16-lane half (wave32)
#pragma unroll
  for (int m = 1; m < 16; m <<= 1) {
#pragma unroll
    for (int r = 0; r < 8; ++r) acc[r] += __shfl_xor(acc[r], m, 32);
  }

  __shared__ float sAcc[8][16];
  if (ln == 0) {
#pragma unroll
    for (int r = 0; r < 8; ++r) sAcc[wave][half * 8 + r] = acc[r];
  }
  __syncthreads();
  if (threadIdx.x < 16) {
    int m = threadIdx.x;
    float s = 0.f;
#pragma unroll
    for (int w = 0; w < 8; ++w) s += sAcc[w][m];
    int jj = j0 + m;
    if (jj < NVERT) {
      float wn = s * INV_2PI;  // sum(2*atan2)/(4*pi)
      outExt[(size_t)b * NVERT + jj] = (wn <= 0.99f) ? 1.0f : 0.0f;
    }
  }
}

// ---------------------------------------------------------------------------
// Fused masked nearest-neighbor: d^2 tile via one WMMA, running min on d^2,
// sqrt only once per output.  mask element is geomask[i, j] (argmin axis=1).
// ---------------------------------------------------------------------------
__global__ void __launch_bounds__(256)
v2v_kernel(const float* __restrict__ arow, const float* __restrict__ bcol,
           const float* __restrict__ sqv, const unsigned char* __restrict__ geomask,
           float* __restrict__ outMin, float* __restrict__ outCt) {
  const int b = blockIdx.x / NJT;
  const int jt = blockIdx.x % NJT;
  const int j0 = jt * 16;
  const int lane = threadIdx.x & 31;
  const int wave = threadIdx.x >> 5;
  const int half = lane >> 4;
  const int ln = lane & 15;

  int j = iminh(j0 + ln, NVERT - 1);
  v2f afrag =
      *reinterpret_cast<const v2f*>(arow + ((size_t)b * NVERT + j) * 4 + 2 * half);

  v8f sqj;
#pragma unroll
  for (int r = 0; r < 8; ++r)
    sqj[r] = sqv[(size_t)b * NVERT + iminh(j0 + 8 * half + r, NVERT - 1)];

  const float INF = __builtin_inff();
  float minv[8];
#pragma unroll
  for (int r = 0; r < 8; ++r) minv[r] = INF;

  for (int it = wave; it < NJT; it += 8) {
    int i = iminh(it * 16 + ln, NVERT - 1);
    v2f bfrag = *reinterpret_cast<const v2f*>(bcol +
                                              ((size_t)b * NVERT + i) * 4 + 2 * half);
    v8f d2 = wmma4(afrag, bfrag, sqj);  // sq_i + sq_j - 2 vi.vj
    const unsigned char* mrow = geomask + (size_t)i * NVERT;
#pragma unroll
    for (int r = 0; r < 8; ++r) {
      int jj = j0 + 8 * half + r;
      unsigned char mb = (jj < NVERT) ? mrow[jj] : (unsigned char)0;
      if (mb) minv[r] = fminf(minv[r], d2[r]);
    }
  }

#pragma unroll
  for (int m = 1; m < 16; m <<= 1) {
#pragma unroll
    for (int r = 0; r < 8; ++r)
      minv[r] = fminf(minv[r], __shfl_xor(minv[r], m, 32));
  }

  __shared__ float sMin[8][16];
  if (ln == 0) {
#pragma unroll
    for (int r = 0; r < 8; ++r) sMin[wave][half * 8 + r] = minv[r];
  }
  __syncthreads();
  if (threadIdx.x < 16) {
    int m = threadIdx.x;
    float d2m = sMin[0][m];
#pragma unroll
    for (int w = 1; w < 8; ++w) d2m = fminf(d2m, sMin[w][m]);
    int jj = j0 + m;
    if (jj < NVERT) {
      float d = fast_sqrt(fmaxf(d2m, 1e-12f));
      outMin[(size_t)b * NVERT + jj] = d;
      outCt[(size_t)b * NVERT + jj] = (d < EUCLTHRES) ? 1.0f : 0.0f;
    }
  }
}

extern "C" void kernel_launch(void* const* d_in, const int* in_sizes, int n_in,
                              void* d_out, int out_size, void* d_ws, size_t ws_size,
                              hipStream_t stream) {
  (void)in_sizes; (void)n_in; (void)out_size; (void)ws_size;
  const float* verts = (const float*)d_in[0];            // [2, NV, 3] f32
  const int* faces = (const int*)d_in[1];                // [NF, 3] i32
  const unsigned char* geomask = (const unsigned char*)d_in[2];  // [NV, NV] bool
  float* out = (float*)d_out;  // [v2v_min | incontact | exterior], each 2*NV

  float* ws = (float*)d_ws;
  float* arow = ws;                                   // 2*NV*4 floats
  float* bcol = arow + (size_t)NBATCH * NVERT * 4;    // 2*NV*4
  float* sqv = bcol + (size_t)NBATCH * NVERT * 4;     // 2*NV
  float* tri = sqv + (size_t)NBATCH * NVERT;          // 2*NF*28 (16B-aligned)

  int prepN = NBATCH * NFACE;  // covers both vertex (2*NV) and tri (2*NF) work
  hipLaunchKernelGGL(prep_kernel, dim3((prepN + 255) / 256), dim3(256), 0, stream,
                     verts, faces, arow, bcol, sqv, tri);

  dim3 grid(NBATCH * NJT);
  hipLaunchKernelGGL(wn_kernel, grid, dim3(256), 0, stream, arow, sqv, tri,
                     out + 2 * (size_t)NBATCH * NVERT);
  hipLaunchKernelGGL(v2v_kernel, grid, dim3(256), 0, stream, arow, bcol, sqv,
                     geomask, out, out + (size_t)NBATCH * NVERT);
}